// CPQuadUnfoldLayer_50062138802240
// MI455X (gfx1250) — compile-verified
//
#include <hip/hip_runtime.h>

// Problem constants (from reference): B=64, N=4096, D=64, R=8, Q=4 quadrant factors.
#define B_ 64
#define N_ 4096
#define D_ 64
#define R_ 8

typedef __attribute__((ext_vector_type(2))) float v2f;
typedef __attribute__((ext_vector_type(4))) float v4f;
typedef __attribute__((ext_vector_type(8))) float v8f;
typedef __attribute__((ext_vector_type(4))) int   v4i;

typedef __attribute__((address_space(1))) v4i glb_v4i;
typedef __attribute__((address_space(1))) int glb_i32;
typedef __attribute__((address_space(3))) v4i lds_v4i;
typedef __attribute__((address_space(3))) int lds_i32;

// Async global->LDS staging path (gfx1250 GLOBAL_LOAD_ASYNC_TO_LDS_*), guarded so
// the kernel still compiles on toolchains without the builtins.
#if defined(__has_builtin)
#if __has_builtin(__builtin_amdgcn_global_load_async_to_lds_b128) && \
    __has_builtin(__builtin_amdgcn_global_load_async_to_lds_b32)  && \
    __has_builtin(__builtin_amdgcn_s_wait_asynccnt)
#define USE_ASYNC_LDS 1
#endif
#endif
#ifndef USE_ASYNC_LDS
#define USE_ASYNC_LDS 0
#endif

// LDS strides padded to avoid bank conflicts (64 banks x 4B).
#define XS_STRIDE  68   // x tile rows: bank = (4*row + col) % 64 -> conflict-free gathers
#define FIN_STRIDE 68
#define FOT_STRIDE 10   // fout_t rows of 8 floats + pad; keeps b64 loads 8B-aligned
#define LAT_STRIDE 20

__global__ __launch_bounds__(128)
void cp_quad_unfold_kernel(const float* __restrict__ x,
                           const float* __restrict__ f_in,
                           const float* __restrict__ f_tl,
                           const float* __restrict__ f_tr,
                           const float* __restrict__ f_bl,
                           const float* __restrict__ f_br,
                           const float* __restrict__ scale,
                           float* __restrict__ out)
{
    // One workgroup per n. 4 wave32s; wave w owns batch rows [16w, 16w+16).
    __shared__ __align__(16) float xs[B_][XS_STRIDE];          // x[:, n, :]      17408 B
    __shared__ __align__(16) float fin[16][FIN_STRIDE];        // F_in[n] rows (8 real + 8 zero)
    __shared__ __align__(16) float fout_t[4 * D_][FOT_STRIDE]; // fout_t[q*64+o][r]
    __shared__ __align__(16) float lat[4][16][LAT_STRIDE];     // per-wave latents tile
    __shared__ __align__(16) float sscale[16];                 // scale[n, 0:8] + zero pad

    const int n   = blockIdx.x;
    const int tid = threadIdx.x;

    // ---------------- Cooperative staging into LDS ----------------------------
#if USE_ASYNC_LDS
    // x tile: 64 rows x 64 cols, 1024 x 16B direct global->LDS copies
    #pragma unroll
    for (int i = 0; i < 8; ++i) {
        int idx = i * 128 + tid;
        int b   = idx >> 4;
        int d4  = (idx & 15) << 2;
        __builtin_amdgcn_global_load_async_to_lds_b128(
            (glb_v4i*)(x + ((size_t)(b * N_ + n)) * D_ + d4),
            (lds_v4i*)&xs[b][d4], 0, 0);
    }
    // F_in: 8 rows x 64 cols (exactly 128 x 16B)
    {
        int r  = tid >> 4;
        int d4 = (tid & 15) << 2;
        __builtin_amdgcn_global_load_async_to_lds_b128(
            (glb_v4i*)(f_in + ((size_t)n * R_ + r) * D_ + d4),
            (lds_v4i*)&fin[r][d4], 0, 0);
    }
    // Output factors, transposed scatter: fout_t[q*64+o][r] (per-lane LDS dests)
    {
        const float* fq[4] = { f_tl, f_tr, f_bl, f_br };
        #pragma unroll
        for (int q = 0; q < 4; ++q) {
            #pragma unroll
            for (int i = 0; i < 4; ++i) {
                int idx = i * 128 + tid;    // r = idx/64, o = idx%64
                int r = idx >> 6;
                int o = idx & 63;
                __builtin_amdgcn_global_load_async_to_lds_b32(
                    (glb_i32*)(fq[q] + ((size_t)n * R_ + r) * D_ + o),
                    (lds_i32*)&fout_t[q * D_ + o][r], 0, 0);
            }
        }
    }
#else
    #pragma unroll
    for (int i = 0; i < 8; ++i) {
        int idx = i * 128 + tid;
        int b   = idx >> 4;
        int d4  = (idx & 15) << 2;
        v4f v = __builtin_nontemporal_load(
            reinterpret_cast<const v4f*>(x + ((size_t)(b * N_ + n)) * D_ + d4));
        *reinterpret_cast<v4f*>(&xs[b][d4]) = v;
    }
    {
        int r  = tid >> 4;
        int d4 = (tid & 15) << 2;
        v4f v = __builtin_nontemporal_load(
            reinterpret_cast<const v4f*>(f_in + ((size_t)n * R_ + r) * D_ + d4));
        *reinterpret_cast<v4f*>(&fin[r][d4]) = v;
    }
    {
        const float* fq[4] = { f_tl, f_tr, f_bl, f_br };
        #pragma unroll
        for (int q = 0; q < 4; ++q) {
            #pragma unroll
            for (int i = 0; i < 4; ++i) {
                int idx = i * 128 + tid;
                int r = idx >> 6;
                int o = idx & 63;
                float v = __builtin_nontemporal_load(
                    fq[q] + ((size_t)n * R_ + r) * D_ + o);
                fout_t[q * D_ + o][r] = v;
            }
        }
    }
#endif
    // zero pad rows 8..15 of fin (B-matrix cols r=8..15 -> contribute 0)
    #pragma unroll
    for (int i = 0; i < 4; ++i) {
        int idx = i * 128 + tid;            // 512 floats
        fin[8 + (idx >> 6)][idx & 63] = 0.0f;
    }
    // scale row -> LDS, zero-padded to 16 so the hot loop is branch/EXEC-free
    if (tid < 16) {
        sscale[tid] = (tid < R_) ? scale[(size_t)n * R_ + tid] : 0.0f;
    }
#if USE_ASYNC_LDS
    __builtin_amdgcn_s_wait_asynccnt(0);    // all async LDS writes landed
#endif
    __syncthreads();

    const int w    = tid >> 5;       // wave id 0..3
    const int lane = tid & 31;
    const int m    = lane & 15;      // A: row M / B: col N / C: col N
    const int hi   = lane >> 4;      // lane half: K offset +2 (A/B), row offset +8 (C)
    const int b0   = w * 16;

    // ---------------- GEMM1: latents = x_tile(16x64) @ F_in^T(64x16pad) -------
    // v_wmma_f32_16x16x4_f32, 16 K-steps of 4.
    v8f c1 = {};
    #pragma unroll
    for (int s = 0; s < 16; ++s) {
        const int ca = 4 * s + 2 * hi;
        v2f a = *reinterpret_cast<const v2f*>(&xs[b0 + m][ca]);   // A[m][ca..ca+1]
        v2f b = *reinterpret_cast<const v2f*>(&fin[m][ca]);       // B[ca..ca+1][m] = F_in[m][ca..]
        c1 = __builtin_amdgcn_wmma_f32_16x16x4_f32(
                 false, a, false, b, (short)0, c1, false, false);
    }

    // scale per r (= C column N = m); r>=8 pad columns get 0
    {
        const float sc = sscale[m];
        #pragma unroll
        for (int v = 0; v < 8; ++v) c1[v] *= sc;
    }
    // Re-shape C layout -> A layout via wave-private LDS: lat[w][M][r]
    #pragma unroll
    for (int v = 0; v < 8; ++v) lat[w][v + 8 * hi][m] = c1[v];

    // ---------------- GEMM2: out_tile = latents(16x8) @ F_out(8x256) + x ------
    // K is only 8 real -> 2 K-steps. 16 N-tiles cover q in [0,4), o in [0,64).
    #pragma unroll
    for (int t = 0; t < 16; ++t) {
        v8f c2 = {};
        #pragma unroll
        for (int s = 0; s < 2; ++s) {
            const int ka = 4 * s + 2 * hi;                         // r in 0..7
            v2f a  = *reinterpret_cast<const v2f*>(&lat[w][m][ka]);
            v2f bb = *reinterpret_cast<const v2f*>(&fout_t[t * 16 + m][ka]);
            c2 = __builtin_amdgcn_wmma_f32_16x16x4_f32(
                     false, a, false, bb, (short)0, c2, false, false);
        }
        const int q = t >> 2;
        const int o = (t & 3) * 16 + m;
        #pragma unroll
        for (int v = 0; v < 8; ++v) {
            const int brow = b0 + v + 8 * hi;                      // batch row
            const float val = c2[v] + xs[brow][o];                 // residual add
            __builtin_nontemporal_store(
                val, out + ((size_t)(brow * N_ + n) * 4 + q) * D_ + o);
        }
    }
}

extern "C" void kernel_launch(void* const* d_in, const int* in_sizes, int n_in,
                              void* d_out, int out_size, void* d_ws, size_t ws_size,
                              hipStream_t stream) {
    const float* x    = (const float*)d_in[0];
    const float* fin  = (const float*)d_in[1];
    const float* ftl  = (const float*)d_in[2];
    const float* ftr  = (const float*)d_in[3];
    const float* fbl  = (const float*)d_in[4];
    const float* fbr  = (const float*)d_in[5];
    const float* sc   = (const float*)d_in[6];
    float* out = (float*)d_out;

    dim3 grid(N_);
    dim3 block(128);
    hipLaunchKernelGGL(cp_quad_unfold_kernel, grid, block, 0, stream,
                       x, fin, ftl, ftr, fbl, fbr, sc, out);
}